// PositionFeature_47614007443577
// MI455X (gfx1250) — compile-verified
//
#include <hip/hip_runtime.h>

typedef float v2f __attribute__((ext_vector_type(2)));
typedef float v8f __attribute__((ext_vector_type(8)));

constexpr int N_ = 32, S_ = 4, C_ = 128, U_ = 8000, R_ = 4, H_ = 36, W_ = 64;
constexpr int HW_ = H_ * W_;
constexpr int UBLK = 4;              // u's per workgroup == waves per workgroup

__global__ __launch_bounds__(32 * UBLK, 1)
void PositionFeature_kernel(const float* __restrict__ core,
                            const float* __restrict__ grid,
                            const float* __restrict__ feature,
                            const float* __restrict__ biases,
                            float* __restrict__ out)
{
    // sampled channel-pairs: [u_local][c/2][n] -> (chan 2cp, chan 2cp+1)
    __shared__ float2 smp[UBLK][C_ / 2][N_];   // 4*64*32*8 = 64 KB

    const int s    = blockIdx.x / (U_ / UBLK);
    const int u0   = (blockIdx.x % (U_ / UBLK)) * UBLK;
    const int tid  = threadIdx.x;
    const int ul   = tid >> 5;                 // wave id == u_local
    const int lane = tid & 31;
    const int u    = u0 + ul;

    // ---------------- Phase 1: bilinear gather -> LDS (lane owns one n) ---------
    {
        const int n = lane;
        const float2 g = *(const float2*)(grid + (size_t)(n * U_ + u) * 2);
        float gx = fminf(fmaxf((g.x + 1.0f) * (W_ * 0.5f) - 0.5f, 0.0f), (float)(W_ - 1));
        float gy = fminf(fmaxf((g.y + 1.0f) * (H_ * 0.5f) - 0.5f, 0.0f), (float)(H_ - 1));
        // clamp base corner so the 2x2 window is always in-bounds; weight hits 1 at border
        const int xb = min((int)gx, W_ - 2);
        const int yb = min((int)gy, H_ - 2);
        const float wx = gx - (float)xb;
        const float wy = gy - (float)yb;
        const float w00 = (1.0f - wy) * (1.0f - wx);
        const float w01 = (1.0f - wy) * wx;
        const float w10 = wy * (1.0f - wx);
        const float w11 = wy * wx;
        const float* base = core + (size_t)(n * S_ + s) * C_ * HW_ + yb * W_ + xb;
        #pragma unroll 8
        for (int cp = 0; cp < C_ / 2; ++cp) {
            const float* p0 = base + (2 * cp + 0) * HW_;
            const float* p1 = base + (2 * cp + 1) * HW_;
            float a00 = p0[0], a01 = p0[1], a10 = p0[W_], a11 = p0[W_ + 1];
            float b00 = p1[0], b01 = p1[1], b10 = p1[W_], b11 = p1[W_ + 1];
            float2 v;
            v.x = w00 * a00 + w01 * a01 + w10 * a10 + w11 * a11;
            v.y = w00 * b00 + w01 * b01 + w10 * b10 + w11 * b11;
            smp[ul][cp][n] = v;
        }
    }
    __syncthreads();

    // ---------------- Phase 2: D[r,n] = feature(4x128) x sampled(128x32) --------
    {
        const int half = lane >> 4;            // selects K = {0,1} vs {2,3} pair
        const int l16  = lane & 15;            // A: row r ; B/D: column n

        // D row M depends only on A row M; we never store D rows >= R_, so
        // lanes with l16 >= R_ may load any finite row -> clamp, no predication.
        const float* arow = feature + (size_t)(s * U_ + u) * R_ * C_
                          + (size_t)min(l16, R_ - 1) * C_ + 2 * half;

        v8f acc0 = {};                         // n = 0..15
        v8f acc1 = {};                         // n = 16..31
        #pragma unroll
        for (int c0 = 0; c0 < C_; c0 += 4) {
            // A fragment: lane = row r, VGPR v holds K = v + 2*half (contiguous pair)
            v2f a;
            a.x = arow[c0];
            a.y = arow[c0 + 1];
            // B fragment: lane = col n, VGPR v holds K = v + 2*half (channel pair in LDS)
            const int cp = (c0 >> 1) + half;
            float2 b0v = smp[ul][cp][l16];
            float2 b1v = smp[ul][cp][l16 + 16];
            v2f b0; b0.x = b0v.x; b0.y = b0v.y;
            v2f b1; b1.x = b1v.x; b1.y = b1v.y;

            acc0 = __builtin_amdgcn_wmma_f32_16x16x4_f32(false, a, false, b0,
                                                         (short)0, acc0, false, false);
            acc1 = __builtin_amdgcn_wmma_f32_16x16x4_f32(false, a, false, b1,
                                                         (short)0, acc1, false, false);
        }

        // D layout: lane l, VGPR v -> row M = v + 8*(l/16), col N = l%16.
        // Valid rows r=0..3 live in lanes 0..15, VGPRs 0..3. r is contiguous in out.
        if (half == 0) {
            const float4 bias = *(const float4*)(biases + (size_t)(s * U_ + u) * R_);
            float4 o0 = make_float4(acc0[0] + bias.x, acc0[1] + bias.y,
                                    acc0[2] + bias.z, acc0[3] + bias.w);
            float4 o1 = make_float4(acc1[0] + bias.x, acc1[1] + bias.y,
                                    acc1[2] + bias.z, acc1[3] + bias.w);
            const int n0 = l16;
            *(float4*)(out + ((size_t)(n0 * S_ + s) * U_ + u) * R_) = o0;
            *(float4*)(out + ((size_t)((n0 + 16) * S_ + s) * U_ + u) * R_) = o1;
        }
    }
}

extern "C" void kernel_launch(void* const* d_in, const int* in_sizes, int n_in,
                              void* d_out, int out_size, void* d_ws, size_t ws_size,
                              hipStream_t stream) {
    const float* core    = (const float*)d_in[0];
    const float* grid    = (const float*)d_in[1];
    const float* feature = (const float*)d_in[2];
    const float* biases  = (const float*)d_in[3];
    float* out = (float*)d_out;

    dim3 g(S_ * (U_ / UBLK));   // 4 * 2000 = 8000 workgroups
    dim3 b(32 * UBLK);          // 128 threads = 4 waves
    hipLaunchKernelGGL(PositionFeature_kernel, g, b, 0, stream,
                       core, grid, feature, biases, out);
    (void)in_sizes; (void)n_in; (void)out_size; (void)d_ws; (void)ws_size;
}